// MSHF_78185584656829
// MI455X (gfx1250) — compile-verified
//
#include <hip/hip_runtime.h>
#include <cstdint>

// Problem dims (fixed by the reference's setup_inputs)
#define B_ 8
#define C_ 64
#define H_ 256
#define W_ 256
// Tiling
#define HT 16            // tile height (pixels)
#define WT 32            // tile width  (pixels)
#define NC 8             // channels per TDM chunk
#define NCHUNK (C_ / NC) // 8 chunks of 8 channels
#define MAXROWS (HT + 6) // 22
#define MAXPITCH (WT + 6)// 38
#define PLANE_MAX (MAXROWS * MAXPITCH) // 836 floats per channel plane

typedef unsigned int       u32;
typedef unsigned long long u64;
typedef u32 v4u __attribute__((ext_vector_type(4)));
typedef u32 v8u __attribute__((ext_vector_type(8)));

// jnp.pad 'reflect' index map (edge not repeated): -1->1, H->H-2
__device__ __forceinline__ int refl(int i, int n) {
  i = (i < 0) ? -i : i;
  i = (i >= n) ? (2 * n - 2 - i) : i;
  return i;
}

// Issue one Tensor-Data-Mover load: a 3D tile (pitch x rows x NC channels)
// from global (row stride W_, plane stride H_*W_) into LDS at byte offset laddr.
// Descriptor bit layout per CDNA5 ISA ch.8 (D# groups 0-3). All operands are
// wave-uniform -> SGPRs. Inline asm (not the builtin) so it works on both
// ROCm-7.2 (5-arg builtin) and clang-23 (6-arg builtin) toolchains.
__device__ __forceinline__ void tdm_load_tile(u64 gaddr, u32 laddr,
                                              int pitch, int rowsN) {
  const u32 big = 0x100000u; // tensor_dim* >> any tile dim: no OOB clamp ever
  v4u g0;
  g0[0] = 1u;                                   // count=1, user mode
  g0[1] = laddr;                                // lds_addr [63:32]
  g0[2] = (u32)gaddr;                           // global_addr low
  g0[3] = ((u32)(gaddr >> 32) & 0x01FFFFFFu)    // global_addr [56:32]
          | (2u << 30);                         // type = 2 ("image")
  const u64 d0s = (u64)W_;                      // row stride (elements)
  const u64 d1s = (u64)H_ * (u64)W_;            // channel-plane stride
  v8u g1;
  g1[0] = (2u << 16);                                   // data_size=4B, no mask
  g1[1] = (big & 0xFFFFu) << 16;                        // tensor_dim0 lo16
  g1[2] = (big >> 16) | ((big & 0xFFFFu) << 16);        // td0 hi / td1 lo
  g1[3] = (big >> 16) | ((u32)pitch << 16);             // td1 hi | tile_dim0
  g1[4] = (u32)rowsN | ((u32)NC << 16);                 // tile_dim1 | tile_dim2
  g1[5] = (u32)d0s;                                     // dim0_stride[31:0]
  g1[6] = (u32)((d0s >> 32) & 0xFFFFu)
          | ((u32)(d1s & 0xFFFFu) << 16);               // d0s hi16 | d1s lo16
  g1[7] = (u32)(d1s >> 16);                             // d1s [47:16]
  v4u g2;
  g2[0] = big;  // tensor_dim2 (>= NC)
  g2[1] = 0u;   // tensor_dim3 (tile_dim3=0 -> unused)
  g2[2] = 0u;   // tensor_dim2_stride
  g2[3] = 0u;   // tile_dim3 = 0
  v4u g3;
  g3[0] = 0u; g3[1] = 0u; g3[2] = 0u; g3[3] = 0u;
  asm volatile("tensor_load_to_lds %0, %1, %2, %3"
               :: "s"(g0), "s"(g1), "s"(g2), "s"(g3)
               : "memory");
}

// Accumulate the 3-scale MSHF feature for one pixel of one channel.
// ro: 7 reflected row offsets (already *pitch) for dh=-3..+3 around the pixel.
// co: 7 reflected col offsets for dw=-3..+3.
__device__ __forceinline__ void accum_pixel(const float* __restrict__ Lp,
                                            const int* ro, const int* co,
                                            float* __restrict__ acc) {
  const float c00 = Lp[ro[3] + co[3]];
#pragma unroll
  for (int k = 1; k <= 3; ++k) {
    float hh = Lp[ro[3] + co[3 - k]] + Lp[ro[3] + co[3 + k]] - 2.0f * c00;
    float vv = Lp[ro[3 - k] + co[3]] + Lp[ro[3 + k] + co[3]] - 2.0f * c00;
    float hv = Lp[ro[3 - k] + co[3 - k]] - Lp[ro[3 - k] + co[3 + k]]
             - Lp[ro[3 + k] + co[3 - k]] + Lp[ro[3 + k] + co[3 + k]];
    // reference computes on x*255; eps is not scale invariant, so scale here
    hh *= 255.0f; vv *= 255.0f; hv *= 255.0f;
    const float d  = hh - vv;
    const float s  = __builtin_sqrtf(d * d + 4.0f * hv * hv + 1e-7f);
    acc[k - 1] += 0.5f * (hh + vv + s);
  }
}

__global__ __launch_bounds__(256)
void mshf_tdm_kernel(const float* __restrict__ x, float* __restrict__ out) {
  // double-buffered staging: 2 x (8 channels x 38x22 floats) = 53.5 KB LDS
  __shared__ float tile[2][NC * PLANE_MAX];

  const int tx = threadIdx.x;
  const int b  = blockIdx.z;
  const int h0 = blockIdx.y * HT;
  const int w0 = blockIdx.x * WT;

  // staged region clamped to the image; all reflected taps land inside it
  const int hBeg = (h0 - 3 < 0) ? 0 : (h0 - 3);
  const int hEnd = (h0 + HT + 3 > H_) ? H_ : (h0 + HT + 3);
  const int wBeg = (w0 - 3 < 0) ? 0 : (w0 - 3);
  const int wEnd = (w0 + WT + 3 > W_) ? W_ : (w0 + WT + 3);
  const int rowsN = hEnd - hBeg;      // 19..22
  const int pitch = wEnd - wBeg;      // 35..38
  const int plane = rowsN * pitch;    // LDS floats per channel

  // each thread owns 2 vertically-adjacent pixels
  const int px = tx & (WT - 1);
  const int py = (tx >> 5) << 1;
  const int gw = w0 + px;
  const int gh = h0 + py;

  // precompute reflected LDS offsets: 8 rows (covers both owned pixels), 7 cols
  int ro[8];
#pragma unroll
  for (int d = 0; d < 8; ++d)
    ro[d] = (refl(gh - 3 + d, H_) - hBeg) * pitch;
  int co[7];
#pragma unroll
  for (int d = 0; d < 7; ++d)
    co[d] = refl(gw - 3 + d, W_) - wBeg;

  float acc[2][3];
#pragma unroll
  for (int p = 0; p < 2; ++p)
#pragma unroll
    for (int s = 0; s < 3; ++s) acc[p][s] = 0.0f;

  // LDS flat addr: low 32 bits of the generic pointer == LDS byte offset
  const u32 ldsBase  = (u32)(uintptr_t)&tile[0][0];
  const u32 bufBytes = (u32)(NC * PLANE_MAX * sizeof(float));
  const u64 gbase = (u64)(uintptr_t)x +
      4ull * ((u64)b * C_ * H_ * W_ + (u64)hBeg * W_ + (u64)wBeg);
  const u64 chunkStep = 4ull * (u64)NC * H_ * W_;

  const bool wave0 = (tx < 32);
  if (wave0) tdm_load_tile(gbase, ldsBase, pitch, rowsN); // prime chunk 0

  for (int ch = 0; ch < NCHUNK; ++ch) {
    if (wave0) {
      if (ch + 1 < NCHUNK) {
        tdm_load_tile(gbase + (u64)(ch + 1) * chunkStep,
                      ldsBase + ((u32)(ch + 1) & 1u) * bufBytes,
                      pitch, rowsN);
        __builtin_amdgcn_s_wait_tensorcnt(1); // chunk ch landed, ch+1 in flight
      } else {
        __builtin_amdgcn_s_wait_tensorcnt(0);
      }
    }
    __syncthreads();                     // publish TDM writes to all waves

    const float* bufp = &tile[ch & 1][0];
#pragma unroll
    for (int cc = 0; cc < NC; ++cc) {
      const float* Lp = bufp + cc * plane;
      accum_pixel(Lp, &ro[0], co, acc[0]);
      accum_pixel(Lp, &ro[1], co, acc[1]);
    }
    __syncthreads();                     // buffer free before TDM reuses it
  }

  // mean over 64 channels, undo the x255, clamp to [0,1]
  const float sc = 1.0f / (255.0f * 64.0f);
#pragma unroll
  for (int p = 0; p < 2; ++p) {
#pragma unroll
    for (int s = 0; s < 3; ++s) {
      float v = acc[p][s] * sc;
      v = fminf(fmaxf(v, 0.0f), 1.0f);
      out[(((size_t)b * 3 + s) * H_ + (size_t)(gh + p)) * W_ + gw] = v;
    }
  }
}

extern "C" void kernel_launch(void* const* d_in, const int* in_sizes, int n_in,
                              void* d_out, int out_size, void* d_ws, size_t ws_size,
                              hipStream_t stream) {
  (void)in_sizes; (void)n_in; (void)out_size; (void)d_ws; (void)ws_size;
  const float* x = (const float*)d_in[0];
  float* out = (float*)d_out;
  dim3 grid(W_ / WT, H_ / HT, B_);   // (8, 16, 8) = 1024 workgroups
  mshf_tdm_kernel<<<grid, 256, 0, stream>>>(x, out);
}